// MipmappedTexture3DHighPerf_88613765251481
// MI455X (gfx1250) — compile-verified
//
#include <hip/hip_runtime.h>

// ---------------------------------------------------------------------------
// Mipmapped texture sampling, MI455X (gfx1250, wave32).
//   Phase 2 : 4 lanes per query; levels 5..7 staged in LDS (async load-to-LDS)
//   Phase 1a: transpose base CHW -> HWC pyramid level 0 (in d_ws)
//   Phase 1b: build levels 1..7 (each texel = 0.25 * 2x2 box of base)
// k_sample is emitted first so the disasm snippet shows the async-to-LDS path.
// Pyramid texel offsets: off(lvl) = 349520 - (4*w*w - 16)/3, w = 512>>lvl
// Total pyramid: 349,520 texels * 16 ch * 4B = 22,369,280 bytes in d_ws.
// ---------------------------------------------------------------------------

#define NQ          1000000
#define TOTAL_TEX   349520
#define LVL5_OFF    349184          // texel offset of level 5
#define SH_F4       1344            // (16*16 + 8*8 + 4*4) texels * 16ch / 4

#if defined(__HIP_DEVICE_COMPILE__) && \
    __has_builtin(__builtin_amdgcn_global_load_async_to_lds_b128) && \
    __has_builtin(__builtin_amdgcn_s_wait_asynccnt)
#define USE_ASYNC_LDS 1
#else
#define USE_ASYNC_LDS 0
#endif

typedef int v4i __attribute__((ext_vector_type(4)));

__device__ __forceinline__ int imin(int a, int b) { return a < b ? a : b; }

__device__ __forceinline__ int lvl_offset(int lvl) {
  int w = 512 >> lvl;
  return TOTAL_TEX - ((w * w * 4 - 16) / 3);   // exact: divisor of multiple of 3
}

__device__ __forceinline__ float4 wsum4(float4 a, float4 b, float4 c, float4 d,
                                        float wa, float wb, float wc, float wd) {
  float4 r;
  r.x = a.x * wa + b.x * wb + c.x * wc + d.x * wd;
  r.y = a.y * wa + b.y * wb + c.y * wc + d.y * wd;
  r.z = a.z * wa + b.z * wb + c.z * wc + d.z * wd;
  r.w = a.w * wa + b.w * wb + c.w * wc + d.w * wd;
  return r;
}

// ---------------- Phase 2: sample ------------------------------------------
__device__ __forceinline__ float4 sample_level(int lvl, float u, float v, int c4,
                                               const float* __restrict__ pyr,
                                               const float4* sh) {
  int w = 512 >> lvl;
  float fw = (float)(w - 1);
  float x = fminf(fmaxf(u * fw, 0.0f), fw);
  float y = fminf(fmaxf(v * fw, 0.0f), fw);
  int x0 = imin((int)x, w - 1);
  int y0 = imin((int)y, w - 1);
  int x1 = imin(x0 + 1, w - 1);
  int y1 = imin(y0 + 1, w - 1);
  float fx = x - (float)x0;
  float fy = y - (float)y0;
  int off = lvl_offset(lvl);
  // Per-lane LDS-vs-global select -> flat_load_b128 handles aperture per lane.
  const float4* base = (lvl >= 5)
      ? (sh + (off - LVL5_OFF) * 4)
      : (reinterpret_cast<const float4*>(pyr) + (size_t)off * 4);
  float4 v00 = base[(y0 * w + x0) * 4 + c4];
  float4 v01 = base[(y0 * w + x1) * 4 + c4];
  float4 v10 = base[(y1 * w + x0) * 4 + c4];
  float4 v11 = base[(y1 * w + x1) * 4 + c4];
  return wsum4(v00, v01, v10, v11,
               (1.0f - fx) * (1.0f - fy), fx * (1.0f - fy),
               (1.0f - fx) * fy,          fx * fy);
}

__global__ void __launch_bounds__(256)
k_sample(const float2* __restrict__ uv, const float* __restrict__ p,
         const float* __restrict__ pyr, float* __restrict__ out) {
  __shared__ float4 sh[SH_F4];                   // levels 5..7, HWC (21 KB)
  {
    const float4* gsrc = reinterpret_cast<const float4*>(pyr) + (size_t)LVL5_OFF * 4;
    for (int i = threadIdx.x; i < SH_F4; i += 256) {
#if USE_ASYNC_LDS
      __builtin_amdgcn_global_load_async_to_lds_b128(
          (v4i*)(gsrc + i), (v4i*)(sh + i), 0, 0);
#else
      sh[i] = gsrc[i];
#endif
    }
#if USE_ASYNC_LDS
    __builtin_amdgcn_s_wait_asynccnt(0);
#endif
    __syncthreads();
  }

  int t = blockIdx.x * 256 + threadIdx.x;        // 4 lanes per query
  int q = t >> 2, c4 = t & 3;
  if (q >= NQ) return;

  float2 xy = uv[q];
  float lf = p[q] * 7.0f;
  int l0 = (int)lf;
  l0 = imin(l0 < 0 ? 0 : l0, 7);
  int l1 = imin(l0 + 1, 7);
  float alpha = lf - (float)l0;

  float4 s0 = sample_level(l0, xy.x, xy.y, c4, pyr, sh);
  float4 s1 = sample_level(l1, xy.x, xy.y, c4, pyr, sh);
  float ia = 1.0f - alpha;
  float4 r;
  r.x = s0.x * ia + s1.x * alpha;
  r.y = s0.y * ia + s1.y * alpha;
  r.z = s0.z * ia + s1.z * alpha;
  r.w = s0.w * ia + s1.w * alpha;
  reinterpret_cast<float4*>(out)[(size_t)q * 4 + c4] = r;
}

// ---------------- Phase 1a: CHW(16,512,512) -> HWC level 0 -----------------
__global__ void __launch_bounds__(256)
k_transpose(const float* __restrict__ tex, float* __restrict__ pyr) {
  int t = blockIdx.x * 256 + threadIdx.x;        // texel id: y*512 + x
  if (t >= 512 * 512) return;
  float v[16];
#pragma unroll
  for (int c = 0; c < 16; ++c) v[c] = tex[c * (512 * 512) + t];
  float4* o = reinterpret_cast<float4*>(pyr) + t * 4;
  o[0] = make_float4(v[0],  v[1],  v[2],  v[3]);
  o[1] = make_float4(v[4],  v[5],  v[6],  v[7]);
  o[2] = make_float4(v[8],  v[9],  v[10], v[11]);
  o[3] = make_float4(v[12], v[13], v[14], v[15]);
}

// ---------------- Phase 1b: levels 1..7 from base (0.25 * 2x2 box) ---------
__global__ void __launch_bounds__(256)
k_mips(float* __restrict__ pyr) {
  int t = blockIdx.x * 256 + threadIdx.x;        // 87376 texels * 4 chan-quads
  if (t >= 87376 * 4) return;
  int idx = t >> 2, c4 = t & 3;
  int lvl = 1, w = 256, rem = idx;
  while (rem >= w * w) { rem -= w * w; w >>= 1; ++lvl; }
  int y = rem / w, x = rem - y * w;
  int half = 1 << (lvl - 1);
  int r0 = (y << lvl) + half - 1;
  int c0 = (x << lvl) + half - 1;
  const float4* b = reinterpret_cast<const float4*>(pyr);   // level 0 HWC
  float4 a00 = b[(r0 * 512 + c0) * 4 + c4];
  float4 a01 = b[(r0 * 512 + c0 + 1) * 4 + c4];
  float4 a10 = b[((r0 + 1) * 512 + c0) * 4 + c4];
  float4 a11 = b[((r0 + 1) * 512 + c0 + 1) * 4 + c4];
  float4 r = wsum4(a00, a01, a10, a11, 0.25f, 0.25f, 0.25f, 0.25f);
  reinterpret_cast<float4*>(pyr)[(lvl_offset(lvl) + y * w + x) * 4 + c4] = r;
}

// ---------------------------------------------------------------------------
extern "C" void kernel_launch(void* const* d_in, const int* in_sizes, int n_in,
                              void* d_out, int out_size, void* d_ws, size_t ws_size,
                              hipStream_t stream) {
  const float*  uv  = (const float*)d_in[0];   // (1e6, 2)
  const float*  p   = (const float*)d_in[1];   // (1e6,)
  const float*  tex = (const float*)d_in[2];   // (1,16,512,512)
  float* out = (float*)d_out;                  // (1e6, 16)
  float* pyr = (float*)d_ws;                   // needs 22,369,280 B

  k_transpose<<<(512 * 512) / 256, 256, 0, stream>>>(tex, pyr);
  k_mips<<<(87376 * 4 + 255) / 256, 256, 0, stream>>>(pyr);
  k_sample<<<(NQ * 4) / 256, 256, 0, stream>>>(
      (const float2*)uv, p, pyr, out);
}